// GraphConv_35880156791035
// MI455X (gfx1250) — compile-verified
//
#include <hip/hip_runtime.h>
#include <hip/hip_bf16.h>

typedef float v2f __attribute__((ext_vector_type(2)));
typedef float v8f __attribute__((ext_vector_type(8)));

#define B_SZ   1024
#define NIN    128
#define NOUT   128
#define N_SZ   512
#define EPSV   1e-5f

// ---------------------------------------------------------------------------
// K1: s[b,i] = sum_n mask(A[b,n]) * x[b,i,n]
// One block per b (1024 blocks, 256 threads = 8 waves).
// Mask converted to float once into LDS; each wave reduces one 2KB row of x.
// Bandwidth-bound: streams 256 MB of x -> ~11us at 23.3 TB/s.
// ---------------------------------------------------------------------------
__global__ __launch_bounds__(256) void masked_rowsum_kernel(
    const int* __restrict__ A, const float* __restrict__ x,
    const int* __restrict__ dilation, float* __restrict__ S) {
  __shared__ __align__(16) float maskf[N_SZ];
  const int b = blockIdx.x;
  const int t = threadIdx.x;
  const int dil = dilation[0];

  for (int n = t; n < N_SZ; n += 256) {
    const int av = A[b * N_SZ + n];
    maskf[n] = (av == dil || av == -1) ? 1.0f : 0.0f;
  }
  __syncthreads();

  const int wave = t >> 5;
  const int lane = t & 31;

  for (int i = wave; i < NIN; i += 8) {
    const float* row = x + ((size_t)b * NIN + i) * N_SZ;
    float acc = 0.0f;
#pragma unroll
    for (int j = 0; j < 4; ++j) {
      const int n = j * 128 + lane * 4;   // each wave covers 512B contiguous per j
      const float4 xv = *(const float4*)(row + n);
      const float4 mv = *(const float4*)(&maskf[n]);
      acc = fmaf(xv.x, mv.x, acc);
      acc = fmaf(xv.y, mv.y, acc);
      acc = fmaf(xv.z, mv.z, acc);
      acc = fmaf(xv.w, mv.w, acc);
    }
    // wave32 butterfly reduction
#pragma unroll
    for (int off = 16; off > 0; off >>= 1) acc += __shfl_xor(acc, off, 32);
    if (lane == 0) S[b * NIN + i] = acc;
  }
}

// ---------------------------------------------------------------------------
// K2: H = S (1024x128) * W^T (128x128) + bias, via V_WMMA_F32_16X16X4_F32.
// One wave per 16x16 output tile; K-loop of 32 WMMAs (K=4 each).
// A layout (16x4 f32): lane L holds S[m0 + L%16][k0 + 2*(L/16) + {0,1}]
// B layout (4x16 f32): lane L holds B[k0 + 2*(L/16) + {0,1}][n0 + L%16]
//                      with B[k][n] = W[n][k]  (both are contiguous float2)
// C/D layout: vgpr j, lane L -> H[m0 + j + 8*(L/16)][n0 + L%16]
// ---------------------------------------------------------------------------
__global__ __launch_bounds__(256) void gemm_wmma_kernel(
    const float* __restrict__ S, const float* __restrict__ W,
    const float* __restrict__ bias, float* __restrict__ H) {
  const int lane = threadIdx.x & 31;
  const int wave = threadIdx.x >> 5;
  const int tile = blockIdx.x * 8 + wave;     // 512 tiles total
  const int mt = tile >> 3;                   // 64 M-tiles
  const int nt = tile & 7;                    // 8 N-tiles
  const int m0 = mt * 16;
  const int n0 = nt * 16;

  const int r  = lane & 15;                   // row (A) / col (B,C,D) index
  const int kk = (lane >> 4) * 2;             // K sub-offset: 0 or 2

  const float* aptr = S + (size_t)(m0 + r) * NIN + kk;
  const float* bptr = W + (size_t)(n0 + r) * NIN + kk;

  v8f c = {};
#pragma unroll 4
  for (int k0 = 0; k0 < NIN; k0 += 4) {
    const v2f a = *(const v2f*)(aptr + k0);
    const v2f b = *(const v2f*)(bptr + k0);
    c = __builtin_amdgcn_wmma_f32_16x16x4_f32(
        /*neg_a=*/false, a, /*neg_b=*/false, b,
        /*c_mod=*/(short)0, c, /*reuse_a=*/false, /*reuse_b=*/false);
  }

  const float bv = bias[n0 + r];
  const int mhalf = (lane >> 4) * 8;
#pragma unroll
  for (int j = 0; j < 8; ++j) {
    H[(size_t)(m0 + mhalf + j) * NOUT + n0 + r] = c[j] + bv;
  }
}

// ---------------------------------------------------------------------------
// K3: per-channel batch stats: mean and biased variance over B=1024.
// One block (256 threads) per output channel.
// ---------------------------------------------------------------------------
__global__ __launch_bounds__(256) void bn_stats_kernel(
    const float* __restrict__ H, float* __restrict__ stats) {
  __shared__ float sh1[256];
  __shared__ float sh2[256];
  const int o = blockIdx.x;
  const int t = threadIdx.x;
  float s1 = 0.0f, s2 = 0.0f;
  for (int b = t; b < B_SZ; b += 256) {
    const float v = H[(size_t)b * NOUT + o];
    s1 += v;
    s2 = fmaf(v, v, s2);
  }
  sh1[t] = s1;
  sh2[t] = s2;
  __syncthreads();
#pragma unroll
  for (int stride = 128; stride > 0; stride >>= 1) {
    if (t < stride) {
      sh1[t] += sh1[t + stride];
      sh2[t] += sh2[t + stride];
    }
    __syncthreads();
  }
  if (t == 0) {
    const float mean = sh1[0] * (1.0f / (float)B_SZ);
    const float var  = sh2[0] * (1.0f / (float)B_SZ) - mean * mean;
    stats[o]        = mean;
    stats[NOUT + o] = var;
  }
}

// ---------------------------------------------------------------------------
// K4: features = relu((h - mean) * rsqrt(var + eps) * gamma + beta)
// ---------------------------------------------------------------------------
__global__ __launch_bounds__(256) void bn_apply_kernel(
    const float* __restrict__ H, const float* __restrict__ stats,
    const float* __restrict__ gamma, const float* __restrict__ beta,
    float* __restrict__ out) {
  const int idx = blockIdx.x * 256 + threadIdx.x;   // 131072 total
  const int o = idx & (NOUT - 1);
  const float mean = stats[o];
  const float var  = stats[NOUT + o];
  const float v = (H[idx] - mean) * rsqrtf(var + EPSV) * gamma[o] + beta[o];
  out[idx] = fmaxf(v, 0.0f);
}

// ---------------------------------------------------------------------------
// Launcher. Inputs: A, x, W, bias, gamma, beta, dilation.
// Output tuple (A, features): A bits in d_out[0 .. B*N), features after.
// Workspace: S @0 (512KB), H @512KB (512KB), stats @1MB (1KB).
// ---------------------------------------------------------------------------
extern "C" void kernel_launch(void* const* d_in, const int* in_sizes, int n_in,
                              void* d_out, int out_size, void* d_ws, size_t ws_size,
                              hipStream_t stream) {
  const int*   A        = (const int*)d_in[0];
  const float* x        = (const float*)d_in[1];
  const float* W        = (const float*)d_in[2];
  const float* bias     = (const float*)d_in[3];
  const float* gamma    = (const float*)d_in[4];
  const float* beta     = (const float*)d_in[5];
  const int*   dilation = (const int*)d_in[6];

  char* ws = (char*)d_ws;
  float* S     = (float*)(ws);                       // 1024*128 f32
  float* H     = (float*)(ws + (size_t)512 * 1024);  // 1024*128 f32
  float* stats = (float*)(ws + (size_t)1024 * 1024); // 2*128 f32

  float* out        = (float*)d_out;
  float* outFeat    = out + (size_t)B_SZ * N_SZ;     // after the A copy

  // Output[0]: A passthrough (int32 bits into the output buffer)
  hipMemcpyAsync(out, A, (size_t)B_SZ * N_SZ * sizeof(int),
                 hipMemcpyDeviceToDevice, stream);

  // Stage 1: masked reduction over N (bandwidth-bound, 256 MB stream)
  masked_rowsum_kernel<<<B_SZ, 256, 0, stream>>>(A, x, dilation, S);

  // Stage 2: WMMA GEMM + bias
  gemm_wmma_kernel<<<64, 256, 0, stream>>>(S, W, bias, H);

  // Stage 3: batch-norm statistics
  bn_stats_kernel<<<NOUT, 256, 0, stream>>>(H, stats);

  // Stage 4: normalize + affine + relu
  bn_apply_kernel<<<(B_SZ * NOUT) / 256, 256, 0, stream>>>(H, stats, gamma, beta, outFeat);
}